// UpperTraingularSparseLayer_63041529970718
// MI455X (gfx1250) — compile-verified
//
#include <hip/hip_runtime.h>

typedef float v2f __attribute__((ext_vector_type(2)));
typedef float v8f __attribute__((ext_vector_type(8)));

namespace {
constexpr int N     = 16384;
constexpr int WAVES = 4;            // waves per 16-row block (split K-range)
constexpr int BLOCK = 32 * WAVES;   // 128 threads
constexpr int MAX_CHUNK = 4096;     // max K-chunk floats per wave (= 16KB)
}

// Async global->LDS copy of 16B per lane (512B per wave op), ASYNCcnt-tracked.
__device__ __forceinline__ void async_copy_b128(unsigned lds_off,
                                                unsigned long long gaddr) {
  asm volatile("global_load_async_to_lds_b128 %0, %1, off"
               :: "v"(lds_off), "v"(gaddr) : "memory");
}

// y[i] = sum_{k>=i} W[i,k] * x[k]
// One workgroup per 16-row block; 4 waves split the K-range into contiguous
// chunks. W streams straight into WMMA A operands (one b64/lane per k-step,
// layout-exact for V_WMMA_F32_16X16X4_F32). x is staged per-wave into LDS via
// async global->LDS and read as broadcast ds_load_b64 B operands, keeping the
// vmem pipe exclusively for the 512MB upper-triangle W stream.
__global__ __launch_bounds__(BLOCK) void triu_matvec_wmma(
    const float* __restrict__ x, const float* __restrict__ W,
    float* __restrict__ y) {
  __shared__ float xs[WAVES * MAX_CHUNK];   // 64 KB
  __shared__ float red[WAVES][16];

  const int r0    = blockIdx.x << 4;   // first row of this 16-row block
  const int lane  = threadIdx.x & 31;
  const int wave  = threadIdx.x >> 5;
  const int mrow  = lane & 15;         // A row index 0..15
  const int khalf = lane >> 4;         // 0 => K pair {0,1}, 1 => K pair {2,3}

  // K tiles (16 wide) for this block, contiguous per-wave chunks.
  const int tiles = (N - r0) >> 4;
  const int per   = (tiles + WAVES - 1) / WAVES;
  const int tb    = wave * per;
  const int te    = (tb + per < tiles) ? (tb + per) : tiles;
  const int ntile = te - tb;                 // may be <= 0 for tail waves
  const int kb    = r0 + (tb << 4);          // first k of this wave's chunk

  float* xw = xs + wave * MAX_CHUNK;

  // ---- stage this wave's x chunk into LDS (async, own-slot only) ----
  if (ntile > 0) {
    const int bytes = ntile << 6;            // ntile * 16 floats * 4B
    const int ncopy = (bytes + 511) >> 9;    // 512B per wave-wide b128 op
    const unsigned lbase =
        (unsigned)(size_t)xw + (unsigned)(lane << 4);
    const unsigned long long gfirst =
        (unsigned long long)(const void*)x + (unsigned)(kb << 2) +
        (unsigned)(lane << 4);
    const unsigned long long gmax =
        (unsigned long long)(const void*)x + (unsigned)(N * 4 - 16);
    for (int i = 0; i < ncopy; ++i) {
      unsigned long long g = gfirst + (unsigned)(i << 9);
      if (g > gmax) g = gmax;                // tail clamp: stay inside x
      async_copy_b128(lbase + (unsigned)(i << 9), g);
    }
  }
  asm volatile("s_wait_asynccnt 0x0" ::: "memory");

  // A stream pointer: lane holds W[r0+mrow, k + 2*khalf + {0,1}] per b64 load.
  const float* __restrict__ ap =
      W + (size_t)(r0 + mrow) * N + kb + (khalf << 1);
  // B from LDS: lane reads xw-local {x[k+2h], x[k+2h+1]} (broadcast b64).
  const float* __restrict__ bp = xw + (khalf << 1);

  v8f c = {0.f, 0.f, 0.f, 0.f, 0.f, 0.f, 0.f, 0.f};
  int remaining = ntile;

  // Diagonal 16x16 tile (wave 0 only): zero strictly-below-diagonal elements
  // with cndmask selects; branch is wave-uniform so EXEC stays all-1s.
  if (tb == 0 && remaining > 0) {
#pragma unroll
    for (int s = 0; s < 4; ++s) {
      v2f a = *(const v2f*)(ap + (s << 2));
      const int col = (s << 2) + (khalf << 1);
      a.x = (col     >= mrow) ? a.x : 0.f;
      a.y = (col + 1 >= mrow) ? a.y : 0.f;
      v2f b = *(const v2f*)(bp + (s << 2));
      c = __builtin_amdgcn_wmma_f32_16x16x4_f32(false, a, false, b,
                                                (short)0, c, false, false);
    }
    ap += 16; bp += 16; --remaining;
  }

  // Full tiles: 4 b64 W-loads + 4 ds_load_b64 + 4 WMMA per 1KB of W.
  for (int t = 0; t < remaining; ++t) {
#pragma unroll
    for (int s = 0; s < 4; ++s) {
      v2f a = *(const v2f*)(ap + (s << 2));
      v2f b = *(const v2f*)(bp + (s << 2));
      c = __builtin_amdgcn_wmma_f32_16x16x4_f32(false, a, false, b,
                                                (short)0, c, false, false);
    }
    ap += 16; bp += 16;
  }

  // D: VGPR v, lanes 0-15 -> M=v, lanes 16-31 -> M=v+8; all columns identical
  // (broadcast B), so lanes 0/16 hold the 16 partial sums. Cross-wave reduce.
  if (mrow == 0) {
#pragma unroll
    for (int v = 0; v < 8; ++v) red[wave][(khalf << 3) + v] = c[v];
  }
  __syncthreads();
  if (threadIdx.x < 16) {
    float s = 0.f;
#pragma unroll
    for (int w = 0; w < WAVES; ++w) s += red[w][threadIdx.x];
    y[r0 + threadIdx.x] = s;
  }
}

extern "C" void kernel_launch(void* const* d_in, const int* in_sizes, int n_in,
                              void* d_out, int out_size, void* d_ws, size_t ws_size,
                              hipStream_t stream) {
  (void)in_sizes; (void)n_in; (void)out_size; (void)d_ws; (void)ws_size;
  const float* x = (const float*)d_in[0];
  const float* W = (const float*)d_in[1];
  float* y = (float*)d_out;
  dim3 grid(N / 16), block(BLOCK);
  triu_matvec_wmma<<<grid, block, 0, stream>>>(x, W, y);
}